// GIN_DECODER_88012469829886
// MI455X (gfx1250) — compile-verified
//
#include <hip/hip_runtime.h>

#define N_NODES 100000
#define N_EDGES 1600000
#define DIM 128
#define NUM_LAYERS 3

typedef __attribute__((ext_vector_type(2))) float v2f;
typedef __attribute__((ext_vector_type(8))) float v8f;

// ---------------- kernel 1: hbuf = x (coalesced float4 copy) ----------------
__global__ void gin_copy_f4(const float4* __restrict__ in, float4* __restrict__ out, int n4) {
    int i = blockIdx.x * blockDim.x + threadIdx.x;
    if (i < n4) out[i] = in[i];
}

// ---------------- kernel 2: hbuf[dst] += x[src]  (segment_sum) ----------------
// one wave (32 lanes) per edge; each lane moves 4 consecutive floats of the row
__global__ void __launch_bounds__(256)
gin_scatter(const float* __restrict__ x,
            const int* __restrict__ src,
            const int* __restrict__ dst,
            float* __restrict__ hbuf) {
    int e = blockIdx.x * 8 + (threadIdx.x >> 5);
    if (e >= N_EDGES) return;
    int lane = threadIdx.x & 31;
    int s = src[e];
    int d = dst[e];
    float4 v = ((const float4*)(x + (long)s * DIM))[lane];
    float* p = hbuf + (long)d * DIM + lane * 4;
    atomicAdd(p + 0, v.x);
    atomicAdd(p + 1, v.y);
    atomicAdd(p + 2, v.z);
    atomicAdd(p + 3, v.w);
}

// ---------------- kernel 3: fused MLP: out = (relu?)(relu(h@W1+b1)@W2 + b2) ----------------
// block = 16 nodes, 8 waves; each wave owns one 16-wide N tile; K=128 via
// 32 chained v_wmma_f32_16x16x4_f32 accumulations per GEMM.
#define LDS_STRIDE 132  // 128 + 4 pad: fragment reads hit distinct banks

__global__ void __launch_bounds__(256)
gin_mlp(const float* __restrict__ hin,
        const float* __restrict__ W1, const float* __restrict__ b1,
        const float* __restrict__ W2, const float* __restrict__ b2,
        float* __restrict__ out, int relu_out) {
    __shared__ float lds_a[16 * LDS_STRIDE];
    __shared__ float lds_h[16 * LDS_STRIDE];

    const int tid  = threadIdx.x;
    const int wave = tid >> 5;      // 0..7 -> N tile
    const int lane = tid & 31;
    const int l16  = lane & 15;
    const int half = lane >> 4;     // 0: K={k,k+1}, 1: K={k+2,k+3}
    const int n0   = wave << 4;
    const long row0 = (long)blockIdx.x * 16;

    // cooperative load of 16x128 A tile (float4, coalesced). N % 16 == 0 -> no tail.
    for (int i = tid; i < 512; i += 256) {
        int r = i >> 5, c4 = i & 31;
        float4 v = ((const float4*)hin)[(row0 + r) * 32 + c4];
        ((float4*)&lds_a[r * LDS_STRIDE])[c4] = v;
    }
    __syncthreads();

    // ---- GEMM1: acc = A @ W1 ----
    v8f acc = {0.f, 0.f, 0.f, 0.f, 0.f, 0.f, 0.f, 0.f};
#pragma unroll
    for (int k = 0; k < DIM; k += 4) {
        int ka = k + half * 2;
        v2f a, b;
        a.x = lds_a[l16 * LDS_STRIDE + ka];
        a.y = lds_a[l16 * LDS_STRIDE + ka + 1];
        b.x = W1[ka * DIM + n0 + l16];
        b.y = W1[(ka + 1) * DIM + n0 + l16];
        acc = __builtin_amdgcn_wmma_f32_16x16x4_f32(false, a, false, b,
                                                    (short)0, acc, false, false);
    }

    // bias + relu -> hidden tile in LDS (C layout: VGPR r holds M = r + half*8, N = n0+l16)
    float bias1 = b1[n0 + l16];
#pragma unroll
    for (int r = 0; r < 8; ++r) {
        int m = r + half * 8;
        float v = acc[r] + bias1;
        lds_h[m * LDS_STRIDE + n0 + l16] = v > 0.f ? v : 0.f;
    }
    __syncthreads();

    // ---- GEMM2: acc2 = H @ W2 ----
    v8f acc2 = {0.f, 0.f, 0.f, 0.f, 0.f, 0.f, 0.f, 0.f};
#pragma unroll
    for (int k = 0; k < DIM; k += 4) {
        int ka = k + half * 2;
        v2f a, b;
        a.x = lds_h[l16 * LDS_STRIDE + ka];
        a.y = lds_h[l16 * LDS_STRIDE + ka + 1];
        b.x = W2[ka * DIM + n0 + l16];
        b.y = W2[(ka + 1) * DIM + n0 + l16];
        acc2 = __builtin_amdgcn_wmma_f32_16x16x4_f32(false, a, false, b,
                                                     (short)0, acc2, false, false);
    }

    // bias (+ optional relu) -> reuse lds_a as output staging tile
    float bias2 = b2[n0 + l16];
#pragma unroll
    for (int r = 0; r < 8; ++r) {
        int m = r + half * 8;
        float v = acc2[r] + bias2;
        if (relu_out) v = v > 0.f ? v : 0.f;
        lds_a[m * LDS_STRIDE + n0 + l16] = v;
    }
    __syncthreads();

    // coalesced float4 store
    for (int i = tid; i < 512; i += 256) {
        int r = i >> 5, c4 = i & 31;
        ((float4*)out)[(row0 + r) * 32 + c4] = ((float4*)&lds_a[r * LDS_STRIDE])[c4];
    }
}

extern "C" void kernel_launch(void* const* d_in, const int* in_sizes, int n_in,
                              void* d_out, int out_size, void* d_ws, size_t ws_size,
                              hipStream_t stream) {
    const float* x  = (const float*)d_in[0];
    const int*   ei = (const int*)d_in[1];
    const float* W1 = (const float*)d_in[2];
    const float* b1 = (const float*)d_in[3];
    const float* W2 = (const float*)d_in[4];
    const float* b2 = (const float*)d_in[5];
    float* out = (float*)d_out;

    float* hbuf = (float*)d_ws;                     // [N, 128] accumulation buffer
    float* xA   = hbuf + (size_t)N_NODES * DIM;     // ping
    float* xB   = xA   + (size_t)N_NODES * DIM;     // pong

    const int* src = ei;            // edge_index[0]
    const int* dst = ei + N_EDGES;  // edge_index[1]
    const int n4 = N_NODES * DIM / 4;  // 3.2M float4

    const float* xc = x;
    for (int l = 0; l < NUM_LAYERS; ++l) {
        // hbuf = x  (GIN eps=0: h = x + agg, fold x into the accumulator)
        gin_copy_f4<<<(n4 + 255) / 256, 256, 0, stream>>>(
            (const float4*)xc, (float4*)hbuf, n4);
        // hbuf[dst] += x[src]
        gin_scatter<<<N_EDGES / 8, 256, 0, stream>>>(xc, src, dst, hbuf);
        // fused two-GEMM MLP
        float* xo = (l == NUM_LAYERS - 1) ? out : ((l == 0) ? xA : xB);
        gin_mlp<<<N_NODES / 16, 256, 0, stream>>>(
            hbuf,
            W1 + (size_t)l * DIM * DIM, b1 + (size_t)l * DIM,
            W2 + (size_t)l * DIM * DIM, b2 + (size_t)l * DIM,
            xo, (l < NUM_LAYERS - 1) ? 1 : 0);
        xc = xo;
    }
}